// CannyEdgeDetector_77704548319342
// MI455X (gfx1250) — compile-verified
//
#include <hip/hip_runtime.h>

// CDNA5 / gfx1250 fused Canny edge detector.
// HBM floor ~268MB -> ~11.5us @ 23.3 TB/s; 5x5 Gaussian offloaded to
// V_WMMA_F32_16X16X4_F32 (f32 matrix path) so VALU handles only the cheap
// Sobel/NMS tail and the kernel stays memory-bound. Tile origins and NMS
// neighbor offsets are computed arithmetically (no constant-pool loads in
// the WMMA loops).

typedef __attribute__((ext_vector_type(2))) float v2f;
typedef __attribute__((ext_vector_type(8))) float v8f;

#define IMG_H 1024
#define IMG_W 1024
#define OT 64

__device__ __forceinline__ int imin(int a, int b) { return a < b ? a : b; }

__global__ __launch_bounds__(256)
void canny_gfx1250(const float* __restrict__ img, float* __restrict__ out) {
    // LDS, aliased across phase lifetimes:
    //   u1.xs  : quantized input, rows/cols [-4,68) rel  (phases 0-1)
    //   u1.bl  : blurred image,   rows/cols [-2,66) rel  (phases 2-3)
    //   u2.bh  : horizontal blur, rows [-4,68) cols [-2,66) (phases 1-2)
    //   u2.s   : mag + angle bin, rows/cols [-1,65) rel  (phases 3-4)
    __shared__ union {
        float xs[72][73];
        float bl[68][69];
    } u1;
    __shared__ union {
        float bh[72][69];
        struct { float mg[66][67]; unsigned char bn[66][67]; } s;
    } u2;

    const int tid  = threadIdx.x;
    const int lane = tid & 31;
    const int wave = tid >> 5;
    const int R = blockIdx.y * OT;
    const int C = blockIdx.x * OT;
    const float* im = img + (size_t)blockIdx.z * (IMG_H * IMG_W);
    float*       op = out + (size_t)blockIdx.z * (IMG_H * IMG_W);

    // Gaussian(5, sigma=3), normalized (matches cv2.getGaussianKernel(5,3)).
    const float g5[5] = {0.17820324f, 0.21052240f, 0.22254890f,
                         0.21052240f, 0.17820324f};

    // Banded Gaussian weight matrix W[k][i] = g5[k - i] (k in [0,20), i in [0,16)),
    // split into 5 K-chunks of 4. Per the f32 16x4 WMMA operand layout:
    // element (i, k): lane = 16*(k>=2 within chunk) + i, vgpr = k&1.
    // The same register image serves as B in pass 1 and A in pass 2.
    v2f wb[5];
    {
        const int i   = lane & 15;
        const int khi = (lane >> 4) << 1;
#pragma unroll
        for (int c = 0; c < 5; ++c) {
            const int k0 = 4 * c + khi;
            const int d0 = k0 - i, d1 = k0 + 1 - i;
            wb[c][0] = (d0 >= 0 && d0 < 5) ? g5[d0] : 0.0f;
            wb[c][1] = (d1 >= 0 && d1 < 5) ? g5[d1] : 0.0f;
        }
    }

    // ---- Phase 0: load + quantize input with reflect-101 halo (halo=4) ----
    for (unsigned idx = tid; idx < 72u * 72u; idx += 256u) {
        const int r = (int)(idx / 72u), c = (int)(idx % 72u);
        int gr = R - 4 + r, gc = C - 4 + c;
        gr = (gr < 0) ? -gr : ((gr >= IMG_H) ? 2 * IMG_H - 2 - gr : gr);
        gc = (gc < 0) ? -gc : ((gc >= IMG_W) ? 2 * IMG_W - 2 - gc : gc);
        u1.xs[r][c] = floorf(im[gr * IMG_W + gc] * 255.0f);
    }
    __syncthreads();

    // ---- Phase 1: horizontal Gaussian via WMMA ----
    // blurH tiles cover rows [-4,68), cols [-2,66); overlapping (clamped) origins:
    //   tr = -4 + min(16*i, 56), tc = -2 + min(16*j, 52)
    {
        const int m   = lane & 15;
        const int khi = (lane >> 4) << 1;
        for (unsigned t = wave; t < 25u; t += 8u) {   // wave-uniform -> EXEC all ones
            const unsigned ti = t / 5u;
            const unsigned tj = t - 5u * ti;
            const int tr = -4 + imin((int)ti << 4, 56);
            const int tc = -2 + imin((int)tj << 4, 52);
            v8f acc = {};
#pragma unroll
            for (int c = 0; c < 5; ++c) {
                const int k0 = 4 * c + khi;
                v2f a;
                a[0] = u1.xs[tr + m + 4][tc + k0 + 2];       // col = tc-2+k0  (+4 halo)
                a[1] = u1.xs[tr + m + 4][tc + k0 + 3];
                acc = __builtin_amdgcn_wmma_f32_16x16x4_f32(
                    false, a, false, wb[c], (short)0, acc, false, false);
            }
            const int col   = lane & 15;
            const int rbase = (lane >> 4) << 3;              // C/D: rows e / e+8
#pragma unroll
            for (int e = 0; e < 8; ++e)
                u2.bh[tr + rbase + e + 4][tc + col + 2] = acc[e];
        }
    }
    __syncthreads();

    // ---- Phase 2: vertical Gaussian via WMMA (writes bl, aliases xs) ----
    // blur tiles cover rows/cols [-2,66): origin = -2 + min(16*k, 52)
    {
        const int n   = lane & 15;
        const int khi = (lane >> 4) << 1;
        for (unsigned t = wave; t < 25u; t += 8u) {
            const unsigned ti = t / 5u;
            const unsigned tj = t - 5u * ti;
            const int tr = -2 + imin((int)ti << 4, 52);
            const int tc = -2 + imin((int)tj << 4, 52);
            v8f acc = {};
#pragma unroll
            for (int c = 0; c < 5; ++c) {
                const int k0 = 4 * c + khi;
                v2f b;
                b[0] = u2.bh[tr + k0 + 2][tc + n + 2];       // row = tr-2+k0  (+4 halo)
                b[1] = u2.bh[tr + k0 + 3][tc + n + 2];
                acc = __builtin_amdgcn_wmma_f32_16x16x4_f32(
                    false, wb[c], false, b, (short)0, acc, false, false);
            }
            const int col   = lane & 15;
            const int rbase = (lane >> 4) << 3;
#pragma unroll
            for (int e = 0; e < 8; ++e)
                u1.bl[tr + rbase + e + 2][tc + col + 2] = acc[e];
        }
    }
    __syncthreads();

    // ---- Phase 3: Sobel + L1 magnitude + angle bin on [-1,65)^2 ----
    for (unsigned idx = tid; idx < 66u * 66u; idx += 256u) {
        const int r = (int)(idx / 66u) - 1, c = (int)(idx % 66u) - 1;
        const float a00 = u1.bl[r + 1][c + 1], a01 = u1.bl[r + 1][c + 2], a02 = u1.bl[r + 1][c + 3];
        const float a10 = u1.bl[r + 2][c + 1],                            a12 = u1.bl[r + 2][c + 3];
        const float a20 = u1.bl[r + 3][c + 1], a21 = u1.bl[r + 3][c + 2], a22 = u1.bl[r + 3][c + 3];
        const float gx = (a02 + 2.0f * a12 + a22) - (a00 + 2.0f * a10 + a20);
        const float gy = (a20 + 2.0f * a21 + a22) - (a00 + 2.0f * a01 + a02);
        const float adx = fabsf(gx), ady = fabsf(gy);
        unsigned char bin;
        if (ady <= 0.41421356f * adx)       bin = 0;   // ~horizontal grad: E/W
        else if (ady >= 2.41421356f * adx)  bin = 2;   // ~vertical grad:  N/S
        else bin = ((gx > 0.0f) == (gy > 0.0f)) ? 1 : 3;
        const int gr = R + r, gc = C + c;
        const bool inimg = (gr >= 0) && (gr < IMG_H) && (gc >= 0) && (gc < IMG_W);
        u2.s.mg[r + 1][c + 1] = inimg ? (adx + ady) : 0.0f;  // zero-pad for NMS
        u2.s.bn[r + 1][c + 1] = bin;
    }
    __syncthreads();

    // ---- Phase 4: non-max suppression + threshold, write 64x64 outputs ----
    // Neighbor offsets by bin: 0:(0,+1) 1:(-1,+1) 2:(-1,0) 3:(-1,-1), mirrored.
    for (unsigned idx = tid; idx < (unsigned)(OT * OT); idx += 256u) {
        const int r = (int)(idx >> 6), c = (int)(idx & 63u);
        const float m = u2.s.mg[r + 1][c + 1];
        const int   b = u2.s.bn[r + 1][c + 1];
        const int  dy = (b == 0) ? 0 : -1;
        const int  dx = (b == 2) ? 0 : ((b == 3) ? -1 : 1);
        const float n1 = u2.s.mg[r + 1 + dy][c + 1 + dx];
        const float n2 = u2.s.mg[r + 1 - dy][c + 1 - dx];
        const float keep = (m >= n1 && m >= n2 && m > 20.0f) ? 1.0f : 0.0f;
        op[(R + r) * IMG_W + (C + c)] = keep;
    }
}

extern "C" void kernel_launch(void* const* d_in, const int* in_sizes, int n_in,
                              void* d_out, int out_size, void* d_ws, size_t ws_size,
                              hipStream_t stream) {
    (void)n_in; (void)out_size; (void)d_ws; (void)ws_size;
    const float* img = (const float*)d_in[0];
    float* out = (float*)d_out;
    const int batch = in_sizes[0] / (IMG_H * IMG_W);
    dim3 grid(IMG_W / OT, IMG_H / OT, batch);
    canny_gfx1250<<<grid, 256, 0, stream>>>(img, out);
}